// GAT_SRF_83665962926394
// MI455X (gfx1250) — compile-verified
//
#include <hip/hip_runtime.h>

typedef __attribute__((ext_vector_type(2))) float v2f;
typedef __attribute__((ext_vector_type(8))) float v8f;

#define NN 1024
#define FF 64
#define ALPHA 0.2f
#define NEG_BIG -9e15f

// ---------------------------------------------------------------------------
// Kernel 1: three small f32 GEMMs via V_WMMA_F32_16X16X4_F32.
// One wave (32 threads) computes one 16x16 output tile, stepping K by 4.
//   A fragment (16x4 f32, 2 VGPRs): lanes 0-15 -> M=lane,    v0=K0, v1=K1
//                                   lanes16-31 -> M=lane-16, v0=K2, v1=K3
//   B fragment (4x16 f32, 2 VGPRs): mirrored striping (v0: K0/K2, v1: K1/K3,
//                                   N = lane&15)
//   C/D (16x16 f32, 8 VGPRs): VGPR r -> M = r (+8 for high half), N = lane&15
// which = 0: Wh = h @ W            (K=128)
// which = 1: s1 = adj @ a[0:64]    (K=64)
// which = 2: s2 = adj @ a[64:128]  (K=64)
// ---------------------------------------------------------------------------
__global__ __launch_bounds__(32) void gat_gemm_wmma(
    const float* __restrict__ h, const float* __restrict__ adj,
    const float* __restrict__ W, const float* __restrict__ a,
    float* __restrict__ Wh, float* __restrict__ s1, float* __restrict__ s2)
{
  const int bid   = blockIdx.x;
  const int which = bid >> 8;     // 0..2  (256 tiles each: 64 x 4)
  const int t     = bid & 255;
  const int tileM = t >> 2;
  const int tileN = t & 3;

  const float* A;
  const float* B;
  float*       C;
  int          K;
  if (which == 0)      { A = h;   B = W;          C = Wh; K = 2 * FF; }
  else if (which == 1) { A = adj; B = a;          C = s1; K = FF;     }
  else                 { A = adj; B = a + FF*FF;  C = s2; K = FF;     }

  const int  lane = threadIdx.x;
  const int  l15  = lane & 15;
  const bool hi   = lane >= 16;
  const int  m    = tileM * 16 + l15;   // A row for this lane
  const int  n    = tileN * 16 + l15;   // B/C column for this lane

  v8f acc = {};
  for (int k0 = 0; k0 < K; k0 += 4) {
    const int ka = k0 + (hi ? 2 : 0);
    v2f af, bf;
    af.x = A[m * K + ka];
    af.y = A[m * K + ka + 1];
    bf.x = B[ka * FF + n];
    bf.y = B[(ka + 1) * FF + n];
    // 8 args: (neg_a, A, neg_b, B, c_mod, C, reuse_a, reuse_b)
    acc = __builtin_amdgcn_wmma_f32_16x16x4_f32(
        false, af, false, bf, (short)0, acc, false, false);
  }

  const int mbase = tileM * 16 + (hi ? 8 : 0);
#pragma unroll
  for (int r = 0; r < 8; ++r)
    C[(mbase + r) * FF + n] = acc[r];
}

// ---------------------------------------------------------------------------
// Kernel 2: masked leaky-relu softmax over j + h_prime reduction + elu.
// Block = row i, 256 threads: f = tid & 63 (contiguous => coalesced 256B
// transactions), jq = tid >> 6 covers j in strides of 4.
// 3 passes (max / sum+hprime / normalized write); exp is recomputed — the
// kernel is bound by the 268 MB non-temporal attention store, not VALU.
// ---------------------------------------------------------------------------
__global__ __launch_bounds__(256) void gat_softmax_writeback(
    const float* __restrict__ adj, const float* __restrict__ Wh,
    const float* __restrict__ s1, const float* __restrict__ s2,
    float* __restrict__ elu_out, float* __restrict__ att_out)
{
  __shared__ float red[3][4][FF];

  const int i   = blockIdx.x;
  const int tid = threadIdx.x;
  const int f   = tid & (FF - 1);
  const int jq  = tid >> 6;          // 0..3

  const float s1v = s1[i * FF + f];

  // ---- pass 1: column max over j ----
  float m = NEG_BIG;
  for (int j = jq; j < NN; j += 4) {
    const float x   = s1v + s2[j * FF + f];
    const float e   = x > 0.f ? x : ALPHA * x;
    const float val = (adj[j * FF + f] > 0.f) ? e : NEG_BIG;
    m = fmaxf(m, val);
  }
  red[0][jq][f] = m;
  __syncthreads();
  m = fmaxf(fmaxf(red[0][0][f], red[0][1][f]),
            fmaxf(red[0][2][f], red[0][3][f]));

  // ---- pass 2: sum of exp + h_prime accumulation ----
  float s = 0.f, hp = 0.f;
  for (int j = jq; j < NN; j += 4) {
    const float x   = s1v + s2[j * FF + f];
    const float e   = x > 0.f ? x : ALPHA * x;
    const float val = (adj[j * FF + f] > 0.f) ? e : NEG_BIG;
    const float ex  = __expf(val - m);
    s  += ex;
    hp += ex * Wh[j * FF + f];
  }
  red[1][jq][f] = s;
  red[2][jq][f] = hp;
  __syncthreads();
  s  = (red[1][0][f] + red[1][1][f]) + (red[1][2][f] + red[1][3][f]);
  hp = (red[2][0][f] + red[2][1][f]) + (red[2][2][f] + red[2][3][f]);
  const float inv = 1.f / s;

  // ---- pass 3: stream out normalized attention (non-temporal: 268MB > L2) ----
  for (int j = jq; j < NN; j += 4) {
    const float x   = s1v + s2[j * FF + f];
    const float e   = x > 0.f ? x : ALPHA * x;
    const float val = (adj[j * FF + f] > 0.f) ? e : NEG_BIG;
    const float att = __expf(val - m) * inv;
    __builtin_nontemporal_store(att, &att_out[((size_t)i * NN + (size_t)j) * FF + f]);
  }

  if (jq == 0) {
    const float hpv = hp * inv;
    elu_out[i * FF + f] = hpv > 0.f ? hpv : (__expf(hpv) - 1.f);
  }
}

// ---------------------------------------------------------------------------
// Launch. Inputs (setup_inputs order): h (1024x128), adj (1024x64),
// W (128x64), a (128x64). Output: [elu(h_prime) 1024*64 | attention 1024*1024*64].
// Workspace: Wh | s1 | s2 = 3 * 64K floats = 768 KB.
// ---------------------------------------------------------------------------
extern "C" void kernel_launch(void* const* d_in, const int* in_sizes, int n_in,
                              void* d_out, int out_size, void* d_ws, size_t ws_size,
                              hipStream_t stream) {
  const float* h   = (const float*)d_in[0];
  const float* adj = (const float*)d_in[1];
  const float* W   = (const float*)d_in[2];
  const float* a   = (const float*)d_in[3];

  float* out     = (float*)d_out;
  float* elu_out = out;            // 1024*64
  float* att_out = out + NN * FF;  // 1024*1024*64

  float* ws = (float*)d_ws;
  float* Wh = ws;
  float* s1 = ws + NN * FF;
  float* s2 = ws + 2 * NN * FF;

  // 3 GEMMs x 256 tiles, one wave each.
  gat_gemm_wmma<<<dim3(768), dim3(32), 0, stream>>>(h, adj, W, a, Wh, s1, s2);

  // One block per destination row i.
  gat_softmax_writeback<<<dim3(NN), dim3(256), 0, stream>>>(adj, Wh, s1, s2,
                                                            elu_out, att_out);
}